// HBS_38723425140759
// MI455X (gfx1250) — compile-verified
//
#include <hip/hip_runtime.h>

// MI455X (gfx1250) fused PNA graph layer.
// Roofline: ~55 GFLOP fp32, ~0.5 GB HBM traffic (~20us @ 23.3 TB/s); the
// 51 MB projection buffers and <1 MB PNA weights are L2-resident (192 MB L2),
// so the edge gather (~2.5 GB of 512B rows) is L2-latency/bandwidth bound.
// GEMMs use native fp32 WMMA (v_wmma_f32_16x16x4_f32) at full precision.
// Weights are repacked once into (k, k+1) float2 pairs so each WMMA B
// fragment is one global_load_b64 instead of two strided b32 loads.

typedef float v2f __attribute__((ext_vector_type(2)));
typedef float v8f __attribute__((ext_vector_type(8)));

// ---------------------------------------------------------------------------
// Kernel 0: CSR row pointers from sorted rows[] via per-node binary search.
// ---------------------------------------------------------------------------
__global__ void build_row_ptr(const int* __restrict__ rows,
                              int* __restrict__ row_ptr,
                              int n_nodes, int n_edges) {
    int i = blockIdx.x * blockDim.x + threadIdx.x;
    if (i > n_nodes) return;
    int lo = 0, hi = n_edges;
    while (lo < hi) {
        int mid = (lo + hi) >> 1;
        if (rows[mid] < i) lo = mid + 1; else hi = mid;
    }
    row_ptr[i] = lo;
}

// ---------------------------------------------------------------------------
// Kernel 0b: repack a row-major [K][Ncols] weight into [K/2][Ncols] float2
// pairs (src[2r][n], src[2r+1][n]) so the WMMA B fragment (k = 2h+v) is a
// single b64 load per lane.
// ---------------------------------------------------------------------------
__global__ void repack_pairs(const float* __restrict__ src,
                             v2f* __restrict__ dst,
                             int khalf, int ncols) {
    int idx = blockIdx.x * blockDim.x + threadIdx.x;
    if (idx >= khalf * ncols) return;
    int r = idx / ncols;
    int n = idx - r * ncols;
    v2f p;
    p.x = src[(size_t)(2 * r + 0) * ncols + n];
    p.y = src[(size_t)(2 * r + 1) * ncols + n];
    dst[idx] = p;
}

// ---------------------------------------------------------------------------
// Kernel 1: m_a = x @ W_a for a in {0,1,2}.  One block = one 16-row M tile.
// 8 waves; wave w owns N-tile w.  K = 128 -> 32 fp32 WMMAs per weight.
// x tile staged in LDS with stride 132 floats (bank = 4*m mod 64 -> the
// 16-lane ds_load_b64 A-fragment reads are conflict free).
// ---------------------------------------------------------------------------
__global__ __launch_bounds__(256)
void proj_kernel(const float* __restrict__ x,
                 const v2f* __restrict__ wpk,   // [3][64][128] packed pairs
                 float* __restrict__ mbuf, int n_nodes) {
    __shared__ float xs[16 * 132];
    const int row0 = blockIdx.x * 16;
    const int t = threadIdx.x;

    // Stage the contiguous 16x128 x tile (2048 floats) via float4.
    {
        const float4* src = (const float4*)(x + (size_t)row0 * 128);
        #pragma unroll
        for (int q = 0; q < 2; ++q) {
            int i = t + q * 256;          // float4 index, 32 per row
            int r = i >> 5;
            int c = (i & 31) << 2;
            float4 v = src[i];
            xs[r * 132 + c + 0] = v.x;
            xs[r * 132 + c + 1] = v.y;
            xs[r * 132 + c + 2] = v.z;
            xs[r * 132 + c + 3] = v.w;
        }
    }
    __syncthreads();

    const int lane = t & 31;
    const int wave = t >> 5;
    const int m = lane & 15;      // A row
    const int h = lane >> 4;      // lane half -> k pair selector
    const int n = lane & 15;      // D column within N tile
    const int col = wave * 16 + n;

    #pragma unroll
    for (int a = 0; a < 3; ++a) {
        const v2f* Bpk = wpk + (size_t)a * 64 * 128;
        v8f acc = {};
        for (int kk = 0; kk < 32; ++kk) {
            const int kb = kk * 4 + 2 * h;          // k = 2*half + vgpr
            v2f av;
            av.x = xs[m * 132 + kb + 0];
            av.y = xs[m * 132 + kb + 1];
            v2f bv = Bpk[(size_t)(2 * kk + h) * 128 + col];  // one b64 load
            acc = __builtin_amdgcn_wmma_f32_16x16x4_f32(
                false, av, false, bv, (short)0, acc, false, false);
        }
        float* outp = mbuf + (size_t)a * n_nodes * 128;
        #pragma unroll
        for (int v = 0; v < 8; ++v) {   // D: row = v + 8*half, col = n
            outp[(size_t)(row0 + v + 8 * h) * 128 + col] = acc[v];
        }
    }
}

// ---------------------------------------------------------------------------
// Kernel 2: fused gather/segmented-reduce + PNA GEMM.
// Block = 16 nodes, 256 threads (8 wave32).
//  Phase A (per weighting): 16 threads/node, 8 feats/thread; each edge row is
//    one coalesced 512B burst from the L2-resident projection buffer, with a
//    one-edge lookahead global_prefetch_b8 to hide L2 latency.
//    mean|max|min|std written into an LDS tile (stride 516 -> conflict free).
//  Phase B: wave w computes the 16x16 D tile at N-tile w, K=512 -> 128 fp32
//    WMMAs; accumulator persists in VGPRs across all three weightings.
// ---------------------------------------------------------------------------
__global__ __launch_bounds__(256)
void pna_kernel(const float* __restrict__ mbuf,
                const int* __restrict__ row_ptr,
                const int* __restrict__ cols,
                const float* __restrict__ vals_A,
                const float* __restrict__ vals_C,
                const v2f* __restrict__ wpk,       // [3][256][128] packed pairs
                const float* __restrict__ pna_b,   // [3][128]
                float* __restrict__ out, int n_nodes) {
    __shared__ float feats[16 * 516];
    const int node0 = blockIdx.x * 16;
    const int t = threadIdx.x;
    const int lane = t & 31;
    const int wave = t >> 5;

    // Aggregation mapping: 16 threads per node, 8 features per thread.
    const int j   = t >> 4;        // local node 0..15
    const int sub = t & 15;        // feature group (8 floats)
    const int node = node0 + j;
    const int estart = row_ptr[node];
    const int eend   = row_ptr[node + 1];
    const float deg  = (float)(eend - estart);
    const float degc = deg > 0.0f ? deg : 1.0f;
    const bool  has  = deg > 0.0f;

    // WMMA fragment mapping.
    const int fm = lane & 15;      // A row
    const int fh = lane >> 4;      // lane half
    const int fn = lane & 15;      // D column within tile
    const int col = wave * 16 + fn;

    v8f acc = {};                  // 16x16 output tile, persists across a

    for (int a = 0; a < 3; ++a) {
        const float* mp = mbuf + (size_t)a * n_nodes * 128;

        float s[8], sq[8], mx[8], mn[8];
        #pragma unroll
        for (int q = 0; q < 8; ++q) {
            s[q] = 0.0f; sq[q] = 0.0f;
            mx[q] = -__builtin_inff();
            mn[q] =  __builtin_inff();
        }
        for (int e = estart; e < eend; ++e) {
            const int c = cols[e];
            // Lookahead prefetch of the next neighbor's message row (L2 hit
            // path); lowers to global_prefetch_b8, silently dropped if moot.
            {
                int en = e + 1 < eend ? e + 1 : e;
                __builtin_prefetch(mp + (size_t)cols[en] * 128 + sub * 8, 0, 3);
            }
            float val;
            if (a == 0)      val = vals_A[e];
            else if (a == 1) val = vals_C[e];
            else             val = vals_A[e] * vals_C[e];
            const float* rp = mp + (size_t)c * 128 + sub * 8;
            float4 r0 = *(const float4*)(rp);
            float4 r1 = *(const float4*)(rp + 4);
            float g[8] = {r0.x, r0.y, r0.z, r0.w, r1.x, r1.y, r1.z, r1.w};
            #pragma unroll
            for (int q = 0; q < 8; ++q) {
                const float gv = g[q] * val;
                s[q] += gv;
                sq[q] = fmaf(gv, gv, sq[q]);
                mx[q] = fmaxf(mx[q], gv);
                mn[q] = fminf(mn[q], gv);
            }
        }
        #pragma unroll
        for (int q = 0; q < 8; ++q) {
            const float mean = s[q] / degc;
            float var = sq[q] / degc - mean * mean;
            var = var > 0.0f ? var : 0.0f;
            const float sd = sqrtf(var + 1e-5f);
            const int f = sub * 8 + q;
            feats[j * 516 +   0 + f] = mean;
            feats[j * 516 + 128 + f] = has ? mx[q] : 0.0f;
            feats[j * 516 + 256 + f] = has ? mn[q] : 0.0f;
            feats[j * 516 + 384 + f] = sd;
        }
        __syncthreads();

        // GEMM: feats[16x512] @ pna_w[a][512x128]; wave owns N-tile `wave`.
        const v2f* wa = wpk + (size_t)a * 256 * 128;
        for (int kk = 0; kk < 128; ++kk) {
            const int kb = kk * 4 + 2 * fh;
            v2f av;
            av.x = feats[fm * 516 + kb + 0];
            av.y = feats[fm * 516 + kb + 1];
            v2f bv = wa[(size_t)(2 * kk + fh) * 128 + col]; // one b64 load
            acc = __builtin_amdgcn_wmma_f32_16x16x4_f32(
                false, av, false, bv, (short)0, acc, false, false);
        }
        __syncthreads();   // feats tile reused next iteration
    }

    // Summed bias (reference: o1+o2+o3, each with its own bias).
    const float bsum = pna_b[col] + pna_b[128 + col] + pna_b[256 + col];
    #pragma unroll
    for (int v = 0; v < 8; ++v) {
        out[(size_t)(node0 + v + 8 * fh) * 128 + col] = acc[v] + bsum;
    }
}

// ---------------------------------------------------------------------------
extern "C" void kernel_launch(void* const* d_in, const int* in_sizes, int n_in,
                              void* d_out, int out_size, void* d_ws, size_t ws_size,
                              hipStream_t stream) {
    const float* x      = (const float*)d_in[0];
    const int*   rows   = (const int*)  d_in[1];
    const int*   cols   = (const int*)  d_in[2];
    const float* vals_A = (const float*)d_in[3];
    const float* vals_C = (const float*)d_in[4];
    const float* W0     = (const float*)d_in[5];
    const float* W1     = (const float*)d_in[6];
    const float* W2     = (const float*)d_in[7];
    const float* pna_w  = (const float*)d_in[8];
    const float* pna_b  = (const float*)d_in[9];

    const int N = in_sizes[0] / 128;   // 100000 (multiple of 16)
    const int E = in_sizes[1];         // 1600000

    // Workspace layout (8B-aligned sections):
    //   mbuf     : [3][N][128] f32            = 153.6 MB
    //   wpk_proj : [3][64][128]  float2 pairs = 192 KB
    //   wpk_pna  : [3][256][128] float2 pairs = 768 KB
    //   row_ptr  : [N+1] i32
    char* ws = (char*)d_ws;
    float* mbuf     = (float*)ws;                  ws += (size_t)3 * N * 128 * 4;
    v2f*   wpk_proj = (v2f*)ws;                    ws += (size_t)3 * 64 * 128 * 8;
    v2f*   wpk_pna  = (v2f*)ws;                    ws += (size_t)3 * 256 * 128 * 8;
    int*   row_ptr  = (int*)ws;

    build_row_ptr<<<(N + 256) / 256, 256, 0, stream>>>(rows, row_ptr, N, E);

    // One-time weight repacks (tiny).
    repack_pairs<<<(64 * 128 + 255) / 256, 256, 0, stream>>>(W0, wpk_proj + 0 * 64 * 128, 64, 128);
    repack_pairs<<<(64 * 128 + 255) / 256, 256, 0, stream>>>(W1, wpk_proj + 1 * 64 * 128, 64, 128);
    repack_pairs<<<(64 * 128 + 255) / 256, 256, 0, stream>>>(W2, wpk_proj + 2 * 64 * 128, 64, 128);
    repack_pairs<<<(768 * 128 + 255) / 256, 256, 0, stream>>>(pna_w, wpk_pna, 768, 128);

    proj_kernel<<<N / 16, 256, 0, stream>>>(x, wpk_proj, mbuf, N);
    pna_kernel<<<N / 16, 256, 0, stream>>>(mbuf, row_ptr, cols, vals_A, vals_C,
                                           wpk_pna, pna_b, (float*)d_out, N);
}